// ThinPlateSpline_52896817217912
// MI455X (gfx1250) — compile-verified
//
#include <hip/hip_runtime.h>
#include <math.h>

// ThinPlateSpline evaluation for MI455X (gfx1250), wave32 + WMMA f32 16x16x4.
//
// out[q,a] = sum_j k(|u_q - c_j|) * W[j,a] + P(u_q) @ poly
// k(r) = r * ln(r)  (dim==3), 0 where r < 1e-10
//
// Squared distance tile via ONE v_wmma_f32_16x16x4_f32:
//   A[M=cp][K]  = [cx, cy, cz, |c|^2]
//   B[K][N=q]   = [-2ux, -2uy, -2uz, 1]
//   C[M][N]     = |u_q|^2 (broadcast)
//   D = A*B + C = |u|^2 + |c|^2 - 2 u.c
//
// Kernel from sq:  r*ln(r) = sqrt(sq) * (ln2/2) * log2(sq)
//   - ONE v_cmp (sq < EPS2) drives BOTH the clamp select and the zero mask
//   - (ln2/2) folded into the LDS-staged weights (amortized out of hot loop)
//   - raw v_sqrt_f32 / v_log_f32 (domain clamped away from denormals)

typedef __attribute__((ext_vector_type(2))) float v2f;
typedef __attribute__((ext_vector_type(8))) float v8f;

#define BATCH 16384
#define NCP   4096
#define CHUNK 1024
#define BLOCK 256
#define ROWS_PER_BLOCK 128   // 8 waves * 16 queries

#define EPS2      1e-20f          // EPS^2
#define HALF_LN2  0.34657359028f  // 0.5 * ln(2)

__launch_bounds__(BLOCK)
__global__ void tps_wmma_kernel(const float* __restrict__ u,
                                const float* __restrict__ cp,
                                const float* __restrict__ w,
                                const float* __restrict__ poly,
                                float* __restrict__ out)
{
    // Staged tiles: cp -> (x,y,z,|c|^2), w -> (ln2/2)*(w0,w1,w2),0
    __shared__ float4 sCP[CHUNK];
    __shared__ float4 sW[CHUNK];

    const int tid  = threadIdx.x;
    const int wave = tid >> 5;
    const int lane = tid & 31;
    const int half = lane >> 4;   // 0: lanes 0-15, 1: lanes 16-31
    const int q    = lane & 15;   // query column within the 16x16 tile

    const int qIdx = blockIdx.x * ROWS_PER_BLOCK + wave * 16 + q;

    // Query (uniform across the two lane halves for a given q)
    const float ux = u[qIdx * 3 + 0];
    const float uy = u[qIdx * 3 + 1];
    const float uz = u[qIdx * 3 + 2];
    const float usq = ux * ux + uy * uy + uz * uz;

    // B matrix (4x16, loop invariant): VGPR0 = K0|K2 halves, VGPR1 = K1|K3
    v2f bmat;
    bmat.x = half ? (-2.0f * uz) : (-2.0f * ux);
    bmat.y = half ? 1.0f         : (-2.0f * uy);

    // C accumulator input: |u_q|^2 broadcast over all 8 row-VGPRs
    v8f cmat;
#pragma unroll
    for (int k = 0; k < 8; ++k) cmat[k] = usq;

    float acc0 = 0.0f, acc1 = 0.0f, acc2 = 0.0f;

    for (int base = 0; base < NCP; base += CHUNK) {
        __syncthreads();
        // Cooperative stage of this chunk's control points + scaled weights.
        for (int i = tid; i < CHUNK; i += BLOCK) {
            const int g = base + i;
            const float cx = cp[g * 3 + 0];
            const float cy = cp[g * 3 + 1];
            const float cz = cp[g * 3 + 2];
            sCP[i] = make_float4(cx, cy, cz, cx * cx + cy * cy + cz * cz);
            sW[i]  = make_float4(HALF_LN2 * w[g * 3 + 0],
                                 HALF_LN2 * w[g * 3 + 1],
                                 HALF_LN2 * w[g * 3 + 2], 0.0f);
        }
        __syncthreads();

#pragma unroll 2
        for (int cb = 0; cb < CHUNK; cb += 16) {
            // A matrix (16x4): lane picks (x,y) or (z,|c|^2) per half
            const float4 c4 = sCP[cb + q];
            v2f amat;
            amat.x = half ? c4.z : c4.x;
            amat.y = half ? c4.w : c4.y;

            // D[M][q] = squared distance tile
            v8f d = __builtin_amdgcn_wmma_f32_16x16x4_f32(
                false, amat, false, bmat, (short)0, cmat, false, false);

            const int mBase = cb + half * 8;
#pragma unroll
            for (int k = 0; k < 8; ++k) {
                const float sq    = d[k];
                const bool  small = sq < EPS2;       // ONE compare, reused twice
                const float sqc   = small ? EPS2 : sq;
                // (ln2/2 pre-folded into sW)  kv ~ sqrt(sq)*log2(sq)
                float kv = __builtin_amdgcn_sqrtf(sqc) * __builtin_amdgcn_logf(sqc);
                kv = small ? 0.0f : kv;              // r < EPS -> 0
                const float4 w4 = sW[mBase + k];     // half-wave uniform broadcast
                acc0 = fmaf(kv, w4.x, acc0);
                acc1 = fmaf(kv, w4.y, acc1);
                acc2 = fmaf(kv, w4.z, acc2);
            }
        }
    }

    // Lane q (rows 0-7) + lane q+16 (rows 8-15) hold partials for same query.
    acc0 += __shfl_xor(acc0, 16, 32);
    acc1 += __shfl_xor(acc1, 16, 32);
    acc2 += __shfl_xor(acc2, 16, 32);

    if (half == 0) {
#pragma unroll
        for (int a = 0; a < 3; ++a) {
            const float p = poly[0 * 3 + a] + ux * poly[1 * 3 + a]
                          + uy * poly[2 * 3 + a] + uz * poly[3 * 3 + a];
            const float acc = (a == 0) ? acc0 : ((a == 1) ? acc1 : acc2);
            out[qIdx * 3 + a] = acc + p;
        }
    }
}

extern "C" void kernel_launch(void* const* d_in, const int* in_sizes, int n_in,
                              void* d_out, int out_size, void* d_ws, size_t ws_size,
                              hipStream_t stream) {
    const float* u    = (const float*)d_in[0];  // (16384, 3)
    const float* cp   = (const float*)d_in[1];  // (4096, 3)
    const float* w    = (const float*)d_in[2];  // (4096, 3)
    const float* poly = (const float*)d_in[3];  // (4, 3)
    float* out = (float*)d_out;                 // (16384, 3)

    dim3 grid(BATCH / ROWS_PER_BLOCK);          // 128 blocks
    tps_wmma_kernel<<<grid, BLOCK, 0, stream>>>(u, cp, w, poly, out);
}